// InstanceAwareHSICLoss_51256139710651
// MI455X (gfx1250) — compile-verified
//
#include <hip/hip_runtime.h>
#include <hip/hip_bf16.h>
#include <math.h>

// ---------------------------------------------------------------------------
// InstanceAwareHSICLoss for MI455X (gfx1250).
//  - Memory-bound: dominated by 134MB |gate| reduction -> float4 grid-stride.
//  - Per-instance HSIC: one 256-thread block (8 wave32) per instance.
//    * gather + L2-normalize 2x [32,512] fp32 tiles in LDS (~144KB, CDNA5
//      WGP has 320KB LDS so this fits; uses wave32 shuffles for row norms)
//    * Gram = X @ X^T via V_WMMA_F32_16X16X4_F32 (fp32 WMMA, exact precision
//      class of the reference). 4 16x16 tiles per matrix, 8 waves = 8 tiles.
//    * exact torch-median (lower middle, sorted[511] of 1024) via in-LDS
//      bitonic sort, sigma^2 = med + 1e-5, K = exp(-dist/(2 sigma^2))
//    * HSIC = sum((K - rowmean K) * (L - rowmean L)^T) / 961
//  - Deterministic 2-pass reductions (no float atomics).
// ---------------------------------------------------------------------------

typedef __attribute__((ext_vector_type(2))) float v2f;
typedef __attribute__((ext_vector_type(8))) float v8f;

#define DIMS      512
#define NSAMP     32
#define KINST     256
#define XSTR      516           // row pitch in LDS (bank-conflict-free WMMA feeds)
#define GSTR      36            // 32x32 gram pitch
#define NTHREADS  256
#define GATE_BLOCKS 2048

// shared memory layout (float offsets)
#define OFF_XS    0
#define OFF_BS    (NSAMP * XSTR)
#define OFF_GK    (2 * NSAMP * XSTR)
#define OFF_GL    (OFF_GK + NSAMP * GSTR)
#define OFF_SORT  (OFF_GL + NSAMP * GSTR)
#define OFF_DK    (OFF_SORT + 1024)
#define OFF_DL    (OFF_DK + 32)
#define OFF_MK    (OFF_DL + 32)
#define OFF_ML    (OFF_MK + 32)
#define OFF_RED   (OFF_ML + 32)
#define SMEM_FLOATS (OFF_RED + NTHREADS)

// Exact torch.median (lower middle) needs a full sort: bitonic over 1024 floats.
__device__ inline void bitonic_sort_1024(float* s, int tid) {
    for (int k = 2; k <= 1024; k <<= 1) {
        for (int j = k >> 1; j > 0; j >>= 1) {
            __syncthreads();
            for (int i = tid; i < 1024; i += NTHREADS) {
                int ixj = i ^ j;
                if (ixj > i) {
                    float a = s[i], b = s[ixj];
                    bool ascending = ((i & k) == 0);
                    if ((a > b) == ascending) { s[i] = b; s[ixj] = a; }
                }
            }
        }
    }
    __syncthreads();
}

__global__ __launch_bounds__(NTHREADS)
void hsic_kernel(const float* __restrict__ dp,
                 const int* __restrict__ inst_idx,
                 const int* __restrict__ bg_idx,
                 float* __restrict__ hs_out) {
    extern __shared__ float smem[];
    float* xs    = smem + OFF_XS;
    float* bs    = smem + OFF_BS;
    float* gK    = smem + OFF_GK;
    float* gL    = smem + OFF_GL;
    float* sortb = smem + OFF_SORT;
    float* dK    = smem + OFF_DK;
    float* dL    = smem + OFF_DL;
    float* mK    = smem + OFF_MK;
    float* mL    = smem + OFF_ML;
    float* red   = smem + OFF_RED;

    const int tid  = threadIdx.x;
    const int inst = blockIdx.x;
    const int wave = tid >> 5;
    const int lane = tid & 31;

    // ---- gather foreground / background rows into LDS (coalesced per row) ----
    for (int i = tid; i < NSAMP * DIMS; i += NTHREADS) {
        int r = i >> 9;            // DIMS = 512
        int c = i & (DIMS - 1);
        long long rowx = (long long)inst_idx[inst * NSAMP + r];
        long long rowb = (long long)bg_idx[inst * NSAMP + r];
        xs[r * XSTR + c] = dp[rowx * DIMS + c];
        bs[r * XSTR + c] = dp[rowb * DIMS + c];
    }
    __syncthreads();

    // ---- L2 row normalization: each wave owns 4 rows, wave32 butterfly ----
    for (int r = wave; r < NSAMP; r += 8) {
        float sx = 0.f, sb = 0.f;
        for (int c = lane; c < DIMS; c += 32) {
            float vx = xs[r * XSTR + c]; sx += vx * vx;
            float vb = bs[r * XSTR + c]; sb += vb * vb;
        }
        for (int off = 16; off > 0; off >>= 1) {
            sx += __shfl_xor(sx, off, 32);
            sb += __shfl_xor(sb, off, 32);
        }
        float ix = 1.f / fmaxf(sqrtf(sx), 1e-12f);
        float ib = 1.f / fmaxf(sqrtf(sb), 1e-12f);
        for (int c = lane; c < DIMS; c += 32) {
            xs[r * XSTR + c] *= ix;
            bs[r * XSTR + c] *= ib;
        }
    }
    __syncthreads();

    // ---- Gram = X @ X^T via fp32 WMMA 16x16x4 ----------------------------
    // waves 0..3 -> gK tiles (Mt,Nt) in {0,16}^2 ; waves 4..7 -> gL tiles.
    {
        const float* src  = (wave < 4) ? xs : bs;
        float*       gdst = (wave < 4) ? gK : gL;
        int w  = wave & 3;
        int Mt = (w >> 1) << 4;
        int Nt = (w & 1) << 4;
        // A (16x4, MxK): lane%16 = M, K pair base = (lane/16)*2
        // B (4x16, KxN) = X^T     : lane%16 = N, same K pair base -> load row N
        int m  = Mt + (lane & 15);
        int n  = Nt + (lane & 15);
        int kb = (lane >> 4) << 1;
        v8f acc = {};
        for (int k = 0; k < DIMS; k += 4) {
            v2f a = *(const v2f*)&src[m * XSTR + k + kb];
            v2f b = *(const v2f*)&src[n * XSTR + k + kb];
            acc = __builtin_amdgcn_wmma_f32_16x16x4_f32(
                false, a, false, b, (short)0, acc, false, false);
        }
        // C/D layout: VGPR r -> (M = Mt + r + 8*(lane/16), N = Nt + lane%16)
        int mb = Mt + ((lane >> 4) << 3);
        int nn = Nt + (lane & 15);
        for (int r = 0; r < 8; ++r)
            gdst[(mb + r) * GSTR + nn] = acc[r];
    }
    __syncthreads();

    // ---- pairwise distances: d[i][j] = g[i][i] + g[j][j] - 2 g[i][j] -----
    if (tid < 32)            dK[tid] = gK[tid * GSTR + tid];
    else if (tid < 64) { int t = tid - 32; dL[t] = gL[t * GSTR + t]; }
    __syncthreads();

    for (int i = tid; i < 1024; i += NTHREADS) {
        int r = i >> 5, c = i & 31;
        float dk = fmaxf(dK[r] + dK[c] - 2.f * gK[r * GSTR + c], 0.f);
        float dl = fmaxf(dL[r] + dL[c] - 2.f * gL[r * GSTR + c], 0.f);
        gK[r * GSTR + c] = dk;
        gL[r * GSTR + c] = dl;
        sortb[i] = dk;
    }
    __syncthreads();

    // ---- exact medians (lower-middle element, index 511 of 1024) ---------
    bitonic_sort_1024(sortb, tid);
    float medK = sortb[511];
    __syncthreads();
    for (int i = tid; i < 1024; i += NTHREADS)
        sortb[i] = gL[(i >> 5) * GSTR + (i & 31)];
    __syncthreads();
    bitonic_sort_1024(sortb, tid);
    float medL = sortb[511];

    float cKf = 1.f / (2.f * (medK + 1e-5f));
    float cLf = 1.f / (2.f * (medL + 1e-5f));

    // ---- RBF kernels in place --------------------------------------------
    for (int i = tid; i < 1024; i += NTHREADS) {
        int r = i >> 5, c = i & 31;
        gK[r * GSTR + c] = expf(-gK[r * GSTR + c] * cKf);
        gL[r * GSTR + c] = expf(-gL[r * GSTR + c] * cLf);
    }
    __syncthreads();

    // ---- row means --------------------------------------------------------
    if (tid < 32) {
        float s = 0.f;
        for (int j = 0; j < 32; ++j) s += gK[tid * GSTR + j];
        mK[tid] = s * (1.f / 32.f);
    } else if (tid < 64) {
        int t = tid - 32;
        float s = 0.f;
        for (int j = 0; j < 32; ++j) s += gL[t * GSTR + j];
        mL[t] = s * (1.f / 32.f);
    }
    __syncthreads();

    // ---- sum((K - mK) * (L - mL)^T) / (n-1)^2 ----------------------------
    float accum = 0.f;
    for (int i = tid; i < 1024; i += NTHREADS) {
        int r = i >> 5, c = i & 31;
        accum += (gK[r * GSTR + c] - mK[r]) * (gL[c * GSTR + r] - mL[c]);
    }
    red[tid] = accum;
    __syncthreads();
    for (int s = NTHREADS >> 1; s > 0; s >>= 1) {
        if (tid < s) red[tid] += red[tid + s];
        __syncthreads();
    }
    if (tid == 0) hs_out[inst] = red[0] * (1.f / 961.f);   // (32-1)^2
}

// ---------------------------------------------------------------------------
// Pure-bandwidth |gate| partial reduction (deterministic two-pass).
// ---------------------------------------------------------------------------
__global__ __launch_bounds__(NTHREADS)
void gate_l1_kernel(const float* __restrict__ gate,
                    float* __restrict__ partials, long long n) {
    __shared__ float red[NTHREADS];
    float acc = 0.f;
    long long stride = (long long)gridDim.x * blockDim.x * 4;
    for (long long i = ((long long)blockIdx.x * blockDim.x + threadIdx.x) * 4;
         i + 3 < n; i += stride) {
        float4 v = *(const float4*)(gate + i);
        acc += fabsf(v.x) + fabsf(v.y) + fabsf(v.z) + fabsf(v.w);
    }
    red[threadIdx.x] = acc;
    __syncthreads();
    for (int s = NTHREADS >> 1; s > 0; s >>= 1) {
        if (threadIdx.x < s) red[threadIdx.x] += red[threadIdx.x + s];
        __syncthreads();
    }
    if (threadIdx.x == 0) partials[blockIdx.x] = red[0];
}

__global__ __launch_bounds__(NTHREADS)
void finalize_kernel(const float* __restrict__ hs,
                     const float* __restrict__ gp,
                     float* __restrict__ out, long long gate_n) {
    __shared__ float red[NTHREADS];
    int tid = threadIdx.x;

    red[tid] = hs[tid];                    // 256 per-instance HSIC values
    __syncthreads();
    for (int s = NTHREADS >> 1; s > 0; s >>= 1) {
        if (tid < s) red[tid] += red[tid + s];
        __syncthreads();
    }
    float hsum = red[0];                   // valid in tid 0 after barrier
    __syncthreads();

    float g = 0.f;
    for (int i = tid; i < GATE_BLOCKS; i += NTHREADS) g += gp[i];
    red[tid] = g;
    __syncthreads();
    for (int s = NTHREADS >> 1; s > 0; s >>= 1) {
        if (tid < s) red[tid] += red[tid + s];
        __syncthreads();
    }
    if (tid == 0)
        out[0] = hsum * (1.f / (float)KINST)
               + 1e-3f * red[0] / (float)gate_n;
}

extern "C" void kernel_launch(void* const* d_in, const int* in_sizes, int n_in,
                              void* d_out, int out_size, void* d_ws, size_t ws_size,
                              hipStream_t stream) {
    const float* dp       = (const float*)d_in[0];
    const float* gate     = (const float*)d_in[1];
    const int*   inst_idx = (const int*)d_in[2];   // integer inputs -> const int*
    const int*   bg_idx   = (const int*)d_in[3];
    float*       out      = (float*)d_out;
    float*       ws       = (float*)d_ws;

    float* hs = ws;                 // [256]  per-instance HSIC
    float* gp = ws + KINST;         // [2048] gate partial sums
    long long gate_n = (long long)in_sizes[1];

    hipLaunchKernelGGL(gate_l1_kernel, dim3(GATE_BLOCKS), dim3(NTHREADS), 0, stream,
                       gate, gp, gate_n);
    hipLaunchKernelGGL(hsic_kernel, dim3(KINST), dim3(NTHREADS),
                       SMEM_FLOATS * sizeof(float), stream,
                       dp, inst_idx, bg_idx, hs);
    hipLaunchKernelGGL(finalize_kernel, dim3(1), dim3(NTHREADS), 0, stream,
                       hs, gp, out, gate_n);
}